// ObjectCondensationLoss_25726854103375
// MI455X (gfx1250) — compile-verified
//
#include <hip/hip_runtime.h>
#include <hip/hip_bf16.h>
#include <math.h>

#define BB 16
#define NN 131072
#define DD 32
#define SS 512
#define BS (BB*SS)

#define ATT_W  1.0f
#define REP_W  1.5f
#define TAUV   0.7f
#define SUPP_W 0.5f
#define CLAMPV 20.0f
#define EPSV   1e-9f
#define D2MAX  50.0f

typedef __attribute__((ext_vector_type(2))) float v2f;
typedef __attribute__((ext_vector_type(8))) float v8f;

__device__ __forceinline__ float fixnum(float x) {
  if (__builtin_isnan(x)) return 0.0f;
  if (__builtin_isinf(x)) return x > 0.0f ? 3.4028234663852886e38f : -3.4028234663852886e38f;
  return x;
}
// order-preserving float -> uint key (for atomicMax segment-max)
__device__ __forceinline__ unsigned fkey(float f) {
  unsigned u = __float_as_uint(f);
  return (u & 0x80000000u) ? ~u : (u | 0x80000000u);
}
__device__ __forceinline__ float fdec(unsigned k) {
  unsigned u = (k & 0x80000000u) ? (k ^ 0x80000000u) : ~k;
  return __uint_as_float(u);
}

// ---------------- init per-(batch,slice) accumulators ----------------
__global__ void k_init(unsigned* count, unsigned* cpcnt, unsigned* maxkey, int* firstcp,
                       float* denom, float* pcp, float* sump, float* attrs) {
  int i = blockIdx.x * blockDim.x + threadIdx.x;
  if (i >= BS) return;
  count[i] = 0u; cpcnt[i] = 0u; maxkey[i] = 0u; firstcp[i] = NN;
  denom[i] = 0.f; pcp[i] = 0.f; sump[i] = 0.f; attrs[i] = 0.f;
}

// ---------------- pass A: counts, cp counts, first CP, segment max ----------------
__global__ void k_passA(const float* __restrict__ beta, const int* __restrict__ sid,
                        const int* __restrict__ iscp,
                        unsigned* count, unsigned* cpcnt, unsigned* maxkey, int* firstcp) {
  long i = (long)blockIdx.x * blockDim.x + threadIdx.x;
  if (i >= (long)BB * NN) return;
  int b = (int)(i / NN), n = (int)(i % NN);
  int s = sid[i];
  if (s < 0 || s >= SS) return;
  int off = b * SS + s;
  atomicAdd(&count[off], 1u);
  if (iscp[i] == 1) { atomicAdd(&cpcnt[off], 1u); atomicMin(&firstcp[off], n); }
  float bc = fixnum(beta[i]); bc = fminf(fmaxf(bc, -CLAMPV), CLAMPV);
  atomicMax(&maxkey[off], fkey(bc / TAUV));
}

// ---------------- gather CP embeddings (zeros when slice has no CP) ----------------
__global__ void k_gather(const float* __restrict__ embed, const int* __restrict__ firstcp,
                         float* __restrict__ cpvec) {
  int i = blockIdx.x * blockDim.x + threadIdx.x; // over BS*DD
  if (i >= BS * DD) return;
  int d = i % DD; int bs = i / DD; int b = bs / SS;
  int fc = firstcp[bs];
  float v = 0.f;
  if (fc < NN) v = fixnum(embed[((long)b * NN + fc) * DD + d]);
  cpvec[i] = v;
}

// ---------------- pass C: softmax denominators + attraction d^2 (reads embed once) ----------------
__global__ void k_passC(const float* __restrict__ beta, const float* __restrict__ embed,
                        const int* __restrict__ sid, const unsigned* __restrict__ maxkey,
                        float* denom, const float* __restrict__ cpvec, float* attrs) {
  long i = (long)blockIdx.x * blockDim.x + threadIdx.x;
  if (i >= (long)BB * NN) return;
  int b = (int)(i / NN);
  int s = sid[i];
  if (s < 0 || s >= SS) return;
  int off = b * SS + s;
  float bc = fixnum(beta[i]); bc = fminf(fmaxf(bc, -CLAMPV), CLAMPV);
  float mx = fdec(maxkey[off]);
  atomicAdd(&denom[off], expf(bc / TAUV - mx));
  const float* e = embed + i * DD;
  const float* c = cpvec + (long)off * DD;
  float d2 = 0.f;
  #pragma unroll
  for (int d = 0; d < DD; ++d) { float t = fixnum(e[d]) - c[d]; d2 = fmaf(t, t, d2); }
  d2 = fminf(d2, D2MAX);
  atomicAdd(&attrs[off], d2);
}

// ---------------- pass D: p, p_cp, sum_p ----------------
__global__ void k_passD(const float* __restrict__ beta, const int* __restrict__ sid,
                        const int* __restrict__ iscp, const unsigned* __restrict__ maxkey,
                        const float* __restrict__ denom, float* pcp, float* sump) {
  long i = (long)blockIdx.x * blockDim.x + threadIdx.x;
  if (i >= (long)BB * NN) return;
  int b = (int)(i / NN);
  int s = sid[i];
  if (s < 0 || s >= SS) return;
  int off = b * SS + s;
  float bc = fixnum(beta[i]); bc = fminf(fmaxf(bc, -CLAMPV), CLAMPV);
  float e = expf(bc / TAUV - fdec(maxkey[off]));
  float dn = denom[off];
  float p = (dn > 0.f) ? e / dn : 0.f;
  atomicAdd(&sump[off], p);
  if (iscp[i] == 1) atomicAdd(&pcp[off], p);
}

// ---------------- per-batch slice reduction ----------------
__global__ void k_slice(const unsigned* __restrict__ count, const unsigned* __restrict__ cpcnt,
                        const float* __restrict__ pcp, const float* __restrict__ sump,
                        const float* __restrict__ attrs,
                        float* bsum, int* nvalid, float* atot, int* natt, int* cptot) {
  int b = blockIdx.x, t = threadIdx.x;
  __shared__ float rf[2][256];
  __shared__ int   ri[3][256];
  float lb = 0.f, la = 0.f; int lv = 0, ln = 0, lc = 0;
  for (int s = t; s < SS; s += 256) {
    int off = b * SS + s;
    unsigned cnt = count[off], cpc = cpcnt[off];
    lc += (int)cpc;
    bool present = cnt > 0u;
    if (present && cpc == 1u) {
      float pc = pcp[off];
      float ce = -logf(pc + EPSV);
      float supp = (cnt > 1u) ? (sump[off] - pc) / fmaxf((float)cnt - 1.f, 1.f) : 0.f;
      lb += ce + SUPP_W * supp; lv++;
    }
    if (present && cpc > 0u) { la += attrs[off] / fmaxf((float)cnt, 1.f); ln++; }
  }
  rf[0][t] = lb; rf[1][t] = la; ri[0][t] = lv; ri[1][t] = ln; ri[2][t] = lc;
  __syncthreads();
  for (int st = 128; st > 0; st >>= 1) {
    if (t < st) {
      rf[0][t] += rf[0][t + st]; rf[1][t] += rf[1][t + st];
      ri[0][t] += ri[0][t + st]; ri[1][t] += ri[1][t + st]; ri[2][t] += ri[2][t + st];
    }
    __syncthreads();
  }
  if (t == 0) { bsum[b] = rf[0][0]; atot[b] = rf[1][0]; nvalid[b] = ri[0][0]; natt[b] = ri[1][0]; cptot[b] = ri[2][0]; }
}

// ---------------- repulsion: S x S Gram via V_WMMA_F32_16X16X4_F32 ----------------
__global__ void __launch_bounds__(256, 1)
k_rep(const float* __restrict__ cpvec, const int* __restrict__ firstcp, float* __restrict__ repsum) {
  int b = blockIdx.x, t = threadIdx.x;
  __shared__ float cpe[SS * DD];   // 64 KB (fits the 320KB WGP LDS)
  __shared__ float nrm[SS];
  __shared__ float hc[SS];
  __shared__ float red[256];
  for (int i = t; i < SS * DD; i += 256) cpe[i] = cpvec[(long)b * SS * DD + i];
  __syncthreads();
  for (int s = t; s < SS; s += 256) {
    float n = 0.f;
    #pragma unroll
    for (int d = 0; d < DD; ++d) { float v = cpe[s * DD + d]; n = fmaf(v, v, n); }
    nrm[s] = n;
    hc[s] = (firstcp[b * SS + s] < NN) ? 1.f : 0.f;
  }
  __syncthreads();

  int w = t >> 5, lane = t & 31;
  int half = lane >> 4, r = lane & 15;   // A/B frag: lanes 0-15 hold K pair {0,1}, lanes 16-31 hold {2,3}
  float acc = 0.f;
  // 32x32 grid of 16x16 output tiles, 8 waves round-robin
  for (int tile = w; tile < 32 * 32; tile += 8) {
    int ti = tile >> 5, tj = tile & 31;
    int ar = ti * 16 + r, br = tj * 16 + r;
    v8f c; 
    #pragma unroll
    for (int q = 0; q < 8; ++q) c[q] = 0.f;
    #pragma unroll
    for (int kk = 0; kk < 8; ++kk) {           // K = 32 in steps of 4
      int k0 = kk * 4 + half * 2;
      v2f a  = *(const v2f*)&cpe[ar * DD + k0];
      v2f bf = *(const v2f*)&cpe[br * DD + k0]; // Gram: B fragment has same per-lane pattern as A
      c = __builtin_amdgcn_wmma_f32_16x16x4_f32(false, a, false, bf, (short)0, c,
                                                false, false);
    }
    #pragma unroll
    for (int q = 0; q < 8; ++q) {
      int M = q + 8 * half;                     // C/D layout: lanes 16-31 hold M+8 rows
      int I = ti * 16 + M, J = tj * 16 + r;
      if (hc[I] > 0.f && hc[J] > 0.f && I != J) {
        float dd = nrm[I] + nrm[J] - 2.f * c[q];
        dd = fminf(dd, D2MAX);
        acc += expf(-dd);
      }
    }
  }
  red[t] = acc; __syncthreads();
  for (int st = 128; st > 0; st >>= 1) { if (t < st) red[t] += red[t + st]; __syncthreads(); }
  if (t == 0) repsum[b] = red[0];
}

// ---------------- finalize: per-batch losses -> 4 scalars ----------------
__global__ void k_final(const float* bsum, const int* nvalid, const float* atot,
                        const int* natt, const int* cptot, const float* repsum,
                        float* __restrict__ out) {
  if (threadIdx.x != 0 || blockIdx.x != 0) return;
  float sl = 0.f, sb = 0.f, sa = 0.f, sr = 0.f; int vb = 0;
  for (int b = 0; b < BB; ++b) {
    int nv = nvalid[b], na = natt[b];
    float bl = bsum[b] / fmaxf((float)nv, 1.f);
    float at = (na > 0) ? ATT_W * atot[b] / fmaxf((float)na, 1.f) : 0.f;
    long np = (long)na * (long)(na - 1);
    float rp = (np > 0) ? REP_W * repsum[b] / fmaxf((float)np, 1.f) : 0.f;
    bool valid = (cptot[b] > 0) && (nv > 0);
    if (valid) { sl += bl + at + rp; sb += bl; sa += at; sr += rp; vb++; }
  }
  float dv = fmaxf((float)vb, 1.f);
  out[0] = vb > 0 ? sl / dv : 0.f;
  out[1] = vb > 0 ? sb / dv : 0.f;
  out[2] = vb > 0 ? sa / dv : 0.f;
  out[3] = vb > 0 ? sr / dv : 0.f;
}

extern "C" void kernel_launch(void* const* d_in, const int* in_sizes, int n_in,
                              void* d_out, int out_size, void* d_ws, size_t ws_size,
                              hipStream_t stream) {
  const float* beta  = (const float*)d_in[0];
  const float* embed = (const float*)d_in[1];
  const int*   sid   = (const int*)d_in[2];
  const int*   iscp  = (const int*)d_in[3];
  float* out = (float*)d_out;

  char* w = (char*)d_ws;
  unsigned* count   = (unsigned*)w; w += (size_t)BS * 4;
  unsigned* cpcnt   = (unsigned*)w; w += (size_t)BS * 4;
  unsigned* maxkey  = (unsigned*)w; w += (size_t)BS * 4;
  int*      firstcp = (int*)w;      w += (size_t)BS * 4;
  float*    denom   = (float*)w;    w += (size_t)BS * 4;
  float*    pcp     = (float*)w;    w += (size_t)BS * 4;
  float*    sump    = (float*)w;    w += (size_t)BS * 4;
  float*    attrs   = (float*)w;    w += (size_t)BS * 4;
  float*    cpvec   = (float*)w;    w += (size_t)BS * DD * 4;
  float*    bsum    = (float*)w;    w += (size_t)BB * 4;
  float*    atot    = (float*)w;    w += (size_t)BB * 4;
  float*    repsum  = (float*)w;    w += (size_t)BB * 4;
  int*      nvalid  = (int*)w;      w += (size_t)BB * 4;
  int*      natt    = (int*)w;      w += (size_t)BB * 4;
  int*      cptot   = (int*)w;      w += (size_t)BB * 4;

  long BN = (long)BB * NN;
  int pblocks = (int)((BN + 255) / 256);

  k_init<<<(BS + 255) / 256, 256, 0, stream>>>(count, cpcnt, maxkey, firstcp, denom, pcp, sump, attrs);
  k_passA<<<pblocks, 256, 0, stream>>>(beta, sid, iscp, count, cpcnt, maxkey, firstcp);
  k_gather<<<(BS * DD + 255) / 256, 256, 0, stream>>>(embed, firstcp, cpvec);
  k_passC<<<pblocks, 256, 0, stream>>>(beta, embed, sid, maxkey, denom, cpvec, attrs);
  k_passD<<<pblocks, 256, 0, stream>>>(beta, sid, iscp, maxkey, denom, pcp, sump);
  k_slice<<<BB, 256, 0, stream>>>(count, cpcnt, pcp, sump, attrs, bsum, nvalid, atot, natt, cptot);
  k_rep<<<BB, 256, 0, stream>>>(cpvec, firstcp, repsum);
  k_final<<<1, 32, 0, stream>>>(bsum, nvalid, atot, natt, cptot, repsum, out);
}